// DWStateFlowModel_3556232922578
// MI455X (gfx1250) — compile-verified
//
#include <hip/hip_runtime.h>
#include <math.h>

// Problem geometry (fixed by the reference)
#define G_      32768
#define NPG_    32
#define EPG_    256
#define NNODES_ (G_ * NPG_)     // 1,048,576
#define NEDGES_ (G_ * EPG_)     // 8,388,608
#define D_      8
#define L_      3

typedef float v2f __attribute__((ext_vector_type(2)));
typedef float v8f __attribute__((ext_vector_type(8)));

#if defined(__gfx1250__) && __has_builtin(__builtin_amdgcn_global_load_async_to_lds_b32)
#define HAVE_ASYNC_LDS 1
typedef __attribute__((address_space(1))) int* gi1_t;   // global src
typedef __attribute__((address_space(3))) int* li3_t;   // LDS dst
#else
#define HAVE_ASYNC_LDS 0
#endif

__device__ __forceinline__ void wait_async0() {
#if defined(__gfx1250__)
#if __has_builtin(__builtin_amdgcn_s_wait_asynccnt)
    __builtin_amdgcn_s_wait_asynccnt(0);
#else
    asm volatile("s_wait_asynccnt 0" ::: "memory");
#endif
#endif
}

// ---------------------------------------------------------------------------
// Kernel 1: per-feature sum and sum-of-squares over all nodes -> d_ws[0..15]
// ---------------------------------------------------------------------------
__global__ __launch_bounds__(256) void stats_kernel(const float* __restrict__ xin,
                                                    float* __restrict__ acc) {
    __shared__ float red[256 * 16];
    float s[8], q[8];
#pragma unroll
    for (int j = 0; j < 8; ++j) { s[j] = 0.f; q[j] = 0.f; }
    const int stride = gridDim.x * blockDim.x;
    for (int i = blockIdx.x * blockDim.x + threadIdx.x; i < NNODES_; i += stride) {
        const float4* p = (const float4*)(xin + (size_t)i * 8);
        float4 v0 = p[0], v1 = p[1];
        float v[8] = {v0.x, v0.y, v0.z, v0.w, v1.x, v1.y, v1.z, v1.w};
#pragma unroll
        for (int j = 0; j < 8; ++j) { s[j] += v[j]; q[j] += v[j] * v[j]; }
    }
    const int tid = threadIdx.x;
#pragma unroll
    for (int j = 0; j < 8; ++j) { red[tid * 16 + j] = s[j]; red[tid * 16 + 8 + j] = q[j]; }
    __syncthreads();
    for (int off = 128; off > 0; off >>= 1) {
        if (tid < off) {
#pragma unroll
            for (int j = 0; j < 16; ++j) red[tid * 16 + j] += red[(tid + off) * 16 + j];
        }
        __syncthreads();
    }
    if (tid < 16) atomicAdd(&acc[tid], red[tid]);
}

// ---------------------------------------------------------------------------
// Kernel 2: one workgroup (256 threads = 8 wave32) per graph.
// Layer GEMM h = x@W + b runs as V_WMMA_F32_16X16X4_F32 on waves 0/1 with
// pre-padded operands (no exec-masked loads). Input block + attention vectors
// staged via GLOBAL_LOAD_ASYNC_TO_LDS (ASYNCcnt) when available.
// ---------------------------------------------------------------------------
__global__ __launch_bounds__(256) void graph_kernel(
    const float* __restrict__ in_states,
    const float* __restrict__ gat_W,   // (L,8,8)
    const float* __restrict__ gat_asrc,// (L,8)
    const float* __restrict__ gat_adst,// (L,8)
    const float* __restrict__ gat_b,   // (L,8)
    const float* __restrict__ win_Wf,  // (8,8)
    const float* __restrict__ win_ws,  // (8)
    const float* __restrict__ win_b,   // (8)
    const float* __restrict__ atk_Ws,  // (8)
    const float* __restrict__ atk_Wd,  // (8)
    const float* __restrict__ atk_b,   // (1)
    const float* __restrict__ end_Wf,  // (8)
    const float* __restrict__ end_ws,  // (8)
    const float* __restrict__ end_b,   // (1)
    const int*   __restrict__ edges,   // (2,E)
    const float* __restrict__ acc,     // d_ws: 8 sums + 8 sumsq
    float* __restrict__ out)           // [G*8 winner | E attack | G end]
{
    __shared__ float    rawbuf[NPG_ * D_];     // raw in_states block (async target)
    __shared__ float    xbuf[NPG_ * D_];       // normalized / layer input (stride 8)
    __shared__ float    hbuf[NPG_ * 16];       // h = xW+b, padded to 16 cols
    __shared__ float    aggb[NPG_ * D_];       // scatter-aggregate, reused as h8
    __shared__ float    own[NPG_], dice[NPG_];
    __shared__ unsigned segm[NPG_];            // per-dst max (monotone uint)
    __shared__ float    segs[NPG_];            // per-dst sum of exp
    __shared__ float    WlT[L_ * 16 * D_];     // W^T padded: [l][col(16)][k(8)]
    __shared__ float    Blp[L_ * 16];          // bias padded to 16 cols
    __shared__ float    Asrc[L_ * D_], Adst[L_ * D_];
    __shared__ float    scr1[NPG_], scr2[NPG_];
    __shared__ float    redv[4];

    const int tid = threadIdx.x;
    const int g   = blockIdx.x;
    const int f   = tid & 7;

    // --- async stage: input block + attention vectors -> LDS ---
#if HAVE_ASYNC_LDS
    __builtin_amdgcn_global_load_async_to_lds_b32(
        (gi1_t)(in_states + (size_t)g * (NPG_ * D_) + tid),
        (li3_t)&rawbuf[tid], 0, 0);
    if (tid < L_ * D_) {
        __builtin_amdgcn_global_load_async_to_lds_b32(
            (gi1_t)(gat_asrc + tid), (li3_t)&Asrc[tid], 0, 0);
        __builtin_amdgcn_global_load_async_to_lds_b32(
            (gi1_t)(gat_adst + tid), (li3_t)&Adst[tid], 0, 0);
    }
#else
    rawbuf[tid] = in_states[(size_t)g * (NPG_ * D_) + tid];
    if (tid < L_ * D_) { Asrc[tid] = gat_asrc[tid]; Adst[tid] = gat_adst[tid]; }
#endif

    // --- stage pre-padded WMMA operands (transform => plain loads) ---
    for (int i = tid; i < L_ * 16 * D_; i += 256) {
        const int l = i >> 7, r = i & 127, cl = r >> 3, k = r & 7;
        WlT[i] = (cl < 8) ? gat_W[l * 64 + k * 8 + cl] : 0.0f;
    }
    if (tid < L_ * 16) {
        const int l = tid >> 4, cl = tid & 15;
        Blp[tid] = (cl < 8) ? gat_b[l * 8 + cl] : 0.0f;
    }

    // --- this thread's edge, kept in registers for the whole kernel ---
    const int s = edges[(size_t)g * EPG_ + tid] - g * NPG_;
    const int d = edges[(size_t)NEDGES_ + (size_t)g * EPG_ + tid] - g * NPG_;

    // --- normalization constants ---
    const float invN = 1.0f / (float)NNODES_;
    const float mu   = acc[f] * invN;
    const float ex2  = acc[8 + f] * invN;
    float var = ex2 - mu * mu;
    const float sd = sqrtf(var > 0.f ? var : 0.f);

#if HAVE_ASYNC_LDS
    wait_async0();
#endif
    __syncthreads();

    const float raw = rawbuf[tid];
    xbuf[tid] = (raw - mu) / (sd + 1e-5f);
    if (f == 0) own[tid >> 3]  = raw;
    if (f == 1) dice[tid >> 3] = raw;
    __syncthreads();

    const int lane = tid & 31;
    const int wave = tid >> 5;

    for (int l = 0; l < L_; ++l) {
        // ---- h = x @ W_l + b_l via V_WMMA_F32_16X16X4_F32 (waves 0,1) ----
        if (wave < 2) {
            const int col  = lane & 15;            // B/C/D column
            const int koff = (lane >> 4) << 1;     // k-slot: 0 (lanes<16) / 2
            const int row  = col + wave * 16;      // A row
            const float bc = Blp[l * 16 + col];
            v8f c;
#pragma unroll
            for (int v = 0; v < 8; ++v) c[v] = bc;
#pragma unroll
            for (int kt = 0; kt < 2; ++kt) {
                const int k0 = kt * 4 + koff;
                v2f a, b;
                a[0] = xbuf[row * D_ + k0];
                a[1] = xbuf[row * D_ + k0 + 1];
                b[0] = WlT[l * 128 + col * D_ + k0];       // adjacent -> b64
                b[1] = WlT[l * 128 + col * D_ + k0 + 1];
                c = __builtin_amdgcn_wmma_f32_16x16x4_f32(false, a, false, b,
                                                          (short)0, c, false, false);
            }
            const int rbase = wave * 16 + ((lane >> 4) << 3); // lanes>=16: M=v+8
#pragma unroll
            for (int v = 0; v < 8; ++v) hbuf[(rbase + v) * 16 + col] = c[v];
        }
        if (tid < NPG_) { segm[tid] = 0u; segs[tid] = 0.0f; }
        aggb[tid] = 0.0f;
        __syncthreads();

        // ---- per-edge attention logit, leaky_relu(., 0.2) ----
        float e = 0.f;
#pragma unroll
        for (int j = 0; j < D_; ++j)
            e += hbuf[s * 16 + j] * Asrc[l * D_ + j] + hbuf[d * 16 + j] * Adst[l * D_ + j];
        e = (e >= 0.f) ? e : 0.2f * e;

        // segment max via monotone float->uint atomicMax in LDS
        unsigned bits = __float_as_uint(e);
        unsigned key  = (bits & 0x80000000u) ? ~bits : (bits | 0x80000000u);
        atomicMax(&segm[d], key);
        __syncthreads();
        unsigned mk    = segm[d];
        unsigned mbits = (mk & 0x80000000u) ? (mk & 0x7FFFFFFFu) : ~mk;
        const float m  = __uint_as_float(mbits);
        const float ee = __expf(e - m);
        atomicAdd(&segs[d], ee);
        __syncthreads();
        const float alpha = ee / segs[d];
#pragma unroll
        for (int j = 0; j < D_; ++j)
            atomicAdd(&aggb[d * D_ + j], alpha * hbuf[s * 16 + j]);
        __syncthreads();

        // ---- x = elu(agg) ----
        const float z = aggb[tid];
        xbuf[tid] = (z > 0.f) ? z : (__expf(z) - 1.0f);
        __syncthreads();
    }

    // -------------------- heads --------------------
    // h8 = x @ win_Wf (into aggb); pooled-attention logits for both heads
    {
        const int n = tid >> 3;
        float a8 = 0.f;
#pragma unroll
        for (int j = 0; j < D_; ++j) a8 += xbuf[n * D_ + j] * win_Wf[j * 8 + f];
        aggb[tid] = a8;
        if (tid < NPG_) {
            float s1 = 0.f, s2 = 0.f;
#pragma unroll
            for (int j = 0; j < D_; ++j) {
                s1 += xbuf[tid * D_ + j] * win_ws[j];
                s2 += xbuf[tid * D_ + j] * end_ws[j];
            }
            scr1[tid] = s1;
            scr2[tid] = s2;
        }
    }
    __syncthreads();
    if (tid == 0) {
        float m1 = scr1[0], m2 = scr2[0];
        for (int n = 1; n < NPG_; ++n) { m1 = fmaxf(m1, scr1[n]); m2 = fmaxf(m2, scr2[n]); }
        redv[0] = m1; redv[2] = m2;
    }
    __syncthreads();
    if (tid < NPG_) {
        scr1[tid] = __expf(scr1[tid] - redv[0]);
        scr2[tid] = __expf(scr2[tid] - redv[2]);
    }
    __syncthreads();
    if (tid == 0) {
        float s1 = 0.f, s2 = 0.f;
        for (int n = 0; n < NPG_; ++n) { s1 += scr1[n]; s2 += scr2[n]; }
        redv[1] = s1; redv[3] = s2;
    }
    __syncthreads();
    if (tid < 8) {                       // winner_logit[g, tid]
        float o = 0.f;
        for (int n = 0; n < NPG_; ++n) o += scr1[n] * aggb[n * D_ + tid];
        out[(size_t)g * 8 + tid] = o / redv[1] + win_b[tid];
    }
    if (tid == 8) {                      // end_turn_logit[g]
        float o = 0.f;
        for (int n = 0; n < NPG_; ++n) {
            float h1 = 0.f;
#pragma unroll
            for (int j = 0; j < D_; ++j) h1 += xbuf[n * D_ + j] * end_Wf[j];
            o += scr2[n] * h1;
        }
        out[(size_t)G_ * 8 + (size_t)NEDGES_ + g] = o / redv[3] + end_b[0];
    }
    // attack head: one edge per thread, coalesced store
    {
        const bool active = (own[s] != own[d]) && (dice[s] > 1.0f);
        float atk = atk_b[0];
#pragma unroll
        for (int j = 0; j < D_; ++j)
            atk += xbuf[s * D_ + j] * atk_Ws[j] + xbuf[d * D_ + j] * atk_Wd[j];
        out[(size_t)G_ * 8 + (size_t)g * EPG_ + tid] = active ? atk : 0.0f;
    }
}

// ---------------------------------------------------------------------------
extern "C" void kernel_launch(void* const* d_in, const int* in_sizes, int n_in,
                              void* d_out, int out_size, void* d_ws, size_t ws_size,
                              hipStream_t stream) {
    (void)in_sizes; (void)n_in; (void)out_size; (void)ws_size;
    const float* in_states = (const float*)d_in[0];
    const float* gat_W     = (const float*)d_in[1];
    const float* gat_asrc  = (const float*)d_in[2];
    const float* gat_adst  = (const float*)d_in[3];
    const float* gat_b     = (const float*)d_in[4];
    const float* win_Wf    = (const float*)d_in[5];
    const float* win_ws    = (const float*)d_in[6];
    const float* win_b     = (const float*)d_in[7];
    const float* atk_Ws    = (const float*)d_in[8];
    const float* atk_Wd    = (const float*)d_in[9];
    const float* atk_b     = (const float*)d_in[10];
    const float* end_Wf    = (const float*)d_in[11];
    const float* end_ws    = (const float*)d_in[12];
    const float* end_b     = (const float*)d_in[13];
    const int*   edges     = (const int*)d_in[14];
    float*       out       = (float*)d_out;
    float*       acc       = (float*)d_ws;

    (void)hipMemsetAsync(d_ws, 0, 16 * sizeof(float), stream);
    stats_kernel<<<1024, 256, 0, stream>>>(in_states, acc);
    graph_kernel<<<G_, 256, 0, stream>>>(in_states, gat_W, gat_asrc, gat_adst, gat_b,
                                         win_Wf, win_ws, win_b,
                                         atk_Ws, atk_Wd, atk_b,
                                         end_Wf, end_ws, end_b,
                                         edges, acc, out);
}